// CenterWoParamMultiCosineNearLoss_15917148799631
// MI455X (gfx1250) — compile-verified
//
#include <hip/hip_runtime.h>

// Problem constants (match reference)
#define BSZ 8192
#define DD  1024
#define CC  90
#define KK  16

typedef __attribute__((ext_vector_type(2))) float v2f;
typedef __attribute__((ext_vector_type(8))) float v8f;

// ---------------------------------------------------------------- utilities
__global__ void k_zero(int* counts, int* cursor, float* out) {
    int t = threadIdx.x;
    if (t < CC) { counts[t] = 0; cursor[t] = 0; }
    if (t == 0) out[0] = 0.0f;
}

__global__ void k_count(const int* __restrict__ labels, int* __restrict__ counts) {
    int b = blockIdx.x * blockDim.x + threadIdx.x;
    if (b < BSZ) atomicAdd(&counts[labels[b]], 1);
}

__global__ void k_scan(const int* __restrict__ counts, int* __restrict__ offsets,
                       int* __restrict__ cursor) {
    if (threadIdx.x == 0) {
        int acc = 0;
        for (int c = 0; c < CC; ++c) {
            offsets[c] = acc;
            cursor[c]  = acc;
            acc += counts[c];
        }
        offsets[CC] = acc;
    }
}

__global__ void k_scatter(const int* __restrict__ labels, int* __restrict__ cursor,
                          int* __restrict__ sorted) {
    int b = blockIdx.x * blockDim.x + threadIdx.x;
    if (b < BSZ) {
        int pos = atomicAdd(&cursor[labels[b]], 1);
        sorted[pos] = b;
    }
}

// -------------------------------------------------------------- WMMA GEMM
// One wave (32 lanes) per tile: 16 sorted samples of class c  x  16 sub-centers.
// D[16x16] = A[16x1024] * B[1024x16] accumulated via V_WMMA_F32_16X16X4_F32.
__global__ void k_gemm(const float* __restrict__ x, const float* __restrict__ centers,
                       const int* __restrict__ offsets, const int* __restrict__ sorted,
                       float* __restrict__ dlab) {
    const int c   = blockIdx.x;
    const int beg = offsets[c];
    const int end = offsets[c + 1];
    const int row0 = beg + blockIdx.y * 16;
    if (row0 >= end) return;               // uniform branch, EXEC stays full

    const int lane = threadIdx.x;          // 0..31 (wave32)
    const int half = lane >> 4;            // 0 => K={0,1}, 1 => K={2,3}
    const int l16  = lane & 15;

    // A: this lane's sample row (rows past the end duplicate the last row;
    // their results are masked on store).
    int r    = row0 + l16;
    int samp = sorted[(r < end) ? r : (end - 1)];
    const v2f* __restrict__ xp = (const v2f*)(x + (size_t)samp * DD);
    // B: this lane's sub-center row (column l16 of the tile).
    const v2f* __restrict__ bp = (const v2f*)(centers + ((size_t)c * KK + l16) * DD);

    __builtin_prefetch(xp, 0, 1);
    __builtin_prefetch(bp, 0, 1);

    v8f acc = {};
    // k counts float2 units; each iteration covers K-depth of 4 floats.
    #pragma unroll 4
    for (int k = 0; k < DD / 2; k += 2) {
        v2f a = xp[k + half];              // lanes<16: K=0,1  lanes>=16: K=2,3
        v2f b = bp[k + half];
        acc = __builtin_amdgcn_wmma_f32_16x16x4_f32(
            /*neg_a=*/false, a, /*neg_b=*/false, b,
            /*c_mod=*/(short)0, acc, /*reuse_a=*/false, /*reuse_b=*/false);
    }

    // D layout: lane<16 -> rows 0..7 (acc[0..7]) of column l16; lane>=16 -> rows 8..15.
    #pragma unroll
    for (int i = 0; i < 8; ++i) {
        int m  = half * 8 + i;
        int r2 = row0 + m;
        if (r2 < end) {
            int s2 = sorted[r2];
            dlab[(size_t)s2 * KK + l16] = 1.0f - acc[i];
        }
    }
}

// ------------------------------------------------------------- loss reduce
__global__ void k_loss(const float* __restrict__ dlab, float* __restrict__ out) {
    int b = blockIdx.x * blockDim.x + threadIdx.x;
    float loss = 0.0f;
    if (b < BSZ) {
        float d[KK];
        float S = 0.0f, dmin = 3.402823e38f;
        int mi = 0;
        #pragma unroll
        for (int k = 0; k < KK; ++k) {
            d[k] = dlab[b * KK + k];
            S += d[k];
            if (d[k] < dmin) { dmin = d[k]; mi = k; }   // first-min, like argmin
        }
        float t1 = dmin * dmin / S;
        float t2 = 0.0f;
        #pragma unroll
        for (int k = 0; k < KK; ++k)
            if (k != mi) t2 += (1.0f - d[k] / S) * d[k];
        loss = (t1 + t2) * (1.0f / (float)BSZ);
    }
    __shared__ float red[256];
    red[threadIdx.x] = loss;
    __syncthreads();
    for (int s = 128; s > 0; s >>= 1) {
        if (threadIdx.x < (unsigned)s) red[threadIdx.x] += red[threadIdx.x + s];
        __syncthreads();
    }
    if (threadIdx.x == 0) atomicAdd(out, red[0]);
}

// ---------------------------------------------------------------- launcher
extern "C" void kernel_launch(void* const* d_in, const int* in_sizes, int n_in,
                              void* d_out, int out_size, void* d_ws, size_t ws_size,
                              hipStream_t stream) {
    (void)in_sizes; (void)n_in; (void)out_size; (void)ws_size;
    const float* x       = (const float*)d_in[0];
    const int*   labels  = (const int*)d_in[1];
    const float* centers = (const float*)d_in[2];
    float*       out     = (float*)d_out;

    char* ws      = (char*)d_ws;
    int*  counts  = (int*)(ws);                    // 90 ints
    int*  offsets = (int*)(ws + 512);              // 91 ints
    int*  cursor  = (int*)(ws + 1024);             // 90 ints
    int*  sorted  = (int*)(ws + 2048);             // 8192 ints
    float* dlab   = (float*)(ws + 2048 + BSZ * 4); // B*K floats (512 KB)

    k_zero   <<<1, 256, 0, stream>>>(counts, cursor, out);
    k_count  <<<BSZ / 256, 256, 0, stream>>>(labels, counts);
    k_scan   <<<1, 32, 0, stream>>>(counts, offsets, cursor);
    k_scatter<<<BSZ / 256, 256, 0, stream>>>(labels, cursor, sorted);

    dim3 gg(CC, 64);   // 64 row-tiles of 16 covers any plausible class count
    k_gemm   <<<gg, 32, 0, stream>>>(x, centers, offsets, sorted, dlab);

    k_loss   <<<BSZ / 256, 256, 0, stream>>>(dlab, out);
}